// Convolution_15135464751721
// MI455X (gfx1250) — compile-verified
//
#include <hip/hip_runtime.h>
#include <hip/hip_bf16.h>

#define N_NODES 40000
#define N_EDGES 640000

typedef float v2f __attribute__((ext_vector_type(2)));
typedef float v8f __attribute__((ext_vector_type(8)));

__device__ __forceinline__ v8f wmma4(v2f a, v2f b, v8f c) {
  // D = A(16x4 f32) * B(4x16 f32) + C(16x16 f32)
  return __builtin_amdgcn_wmma_f32_16x16x4_f32(false, a, false, b, (short)0, c, false, false);
}
__device__ __forceinline__ v2f ld2(const float* p) { return *(const v2f*)p; }

__device__ __forceinline__ void atomAddF(float* p, float v) {
  unsafeAtomicAdd(p, v);   // global_atomic_add_f32, no return
}

#define F032      0.17677669529663687f   /* 1/sqrt(32) */
#define SILU_G    1.6791f
#define INV_SQ3   0.5773502691896258f
#define INV_SQ2   0.7071067811865476f
#define CC_S      0.3826834323650898f    /* sin(pi/8) */
#define CC_X      0.9238795325112867f    /* cos(pi/8) */
#define INV8      0.125f
#define INV_SQ96  0.10206207261596575f
#define QTR       0.25f                  /* INV_SQRT_NN */

// ---------------------------------------------------------------- zero fill
__global__ void zero_f32(float* __restrict__ p, int n) {
  int i = blockIdx.x * blockDim.x + threadIdx.x;
  if (i < n) p[i] = 0.0f;
}

// ---------------------------------------------------------------- node pre
// l_s = (xs @ W_l10) * f0 * na ; l_v[:,:,i] = (xv[:,:,i] @ W_l11) * f1 * na
__global__ void node_pre(const float* __restrict__ X, const float* __restrict__ NA,
                         const float* __restrict__ Wl10, const float* __restrict__ Wl11,
                         float* __restrict__ Ls, float* __restrict__ Lv) {
  const int lane = threadIdx.x & 31, wv = threadIdx.x >> 5;
  const int tile = blockIdx.x * 8 + wv;
  if (tile * 16 >= N_NODES) return;               // wave-uniform
  const int n = lane & 15, hi = lane >> 4;
  const int arow = tile * 16 + n;
  float nam[8];
#pragma unroll
  for (int r = 0; r < 8; ++r) nam[r] = NA[tile * 16 + 8 * hi + r] * F032;

  { // scalar channel
    v8f c0 = {}, c1 = {};
#pragma unroll
    for (int s = 0; s < 8; ++s) {
      const int kb = 4 * s + 2 * hi;
      v2f a = ld2(X + arow * 128 + kb);
      v2f b0 = { Wl10[kb * 32 + n],      Wl10[(kb + 1) * 32 + n] };
      v2f b1 = { Wl10[kb * 32 + 16 + n], Wl10[(kb + 1) * 32 + 16 + n] };
      c0 = wmma4(a, b0, c0);
      c1 = wmma4(a, b1, c1);
    }
#pragma unroll
    for (int r = 0; r < 8; ++r) {
      const int node = tile * 16 + 8 * hi + r;
      Ls[node * 32 + n]      = c0[r] * nam[r];
      Ls[node * 32 + 16 + n] = c1[r] * nam[r];
    }
  }
#pragma unroll
  for (int i = 0; i < 3; ++i) { // vector channels
    v8f c0 = {}, c1 = {};
#pragma unroll
    for (int s = 0; s < 8; ++s) {
      const int kb = 4 * s + 2 * hi;
      v2f a = { X[arow * 128 + 32 + kb * 3 + i], X[arow * 128 + 32 + (kb + 1) * 3 + i] };
      v2f b0 = { Wl11[kb * 32 + n],      Wl11[(kb + 1) * 32 + n] };
      v2f b1 = { Wl11[kb * 32 + 16 + n], Wl11[(kb + 1) * 32 + 16 + n] };
      c0 = wmma4(a, b0, c0);
      c1 = wmma4(a, b1, c1);
    }
#pragma unroll
    for (int r = 0; r < 8; ++r) {
      const int node = tile * 16 + 8 * hi + r;
      Lv[node * 96 + n * 3 + i]        = c0[r] * nam[r];
      Lv[node * 96 + (16 + n) * 3 + i] = c1[r] * nam[r];
    }
  }
}

// ---------------------------------------------------------------- edge kernel
// 256 threads = 8 waves, 128 edges per block (16 per wave)
__global__ __launch_bounds__(256) void edge_kernel(
    const float* __restrict__ ES,   // (E,16)
    const int*   __restrict__ EIDX, // (2,E)
    const float* __restrict__ EA,   // (E,4)
    const float* __restrict__ Wfc0, // (16,64)
    const float* __restrict__ Wfc1, // (64,160)
    const float* __restrict__ Ls, const float* __restrict__ Lv,
    float* __restrict__ AccS, float* __restrict__ AccV) {
  __shared__ __align__(16) float hbuf[8 * 16 * 68];  // per-wave 16x64 h tile, stride 68
  __shared__ int   sSrc[128], sDst[128];
  __shared__ float sEA[128 * 4];

  const int tid = threadIdx.x;
  const int lane = tid & 31, wv = tid >> 5;
  const int n = lane & 15, hi = lane >> 4;
  const int e0 = blockIdx.x * 128;

  if (tid < 128) {
    sSrc[tid] = EIDX[e0 + tid];
    sDst[tid] = EIDX[N_EDGES + e0 + tid];
  }
  for (int idx = tid; idx < 512; idx += 256) sEA[idx] = EA[e0 * 4 + idx];
  __syncthreads();

  // ---- GEMM1: h = silu(es @ Wfc0 / 4) * gain, with the /8 of w folded in
  const int eg = e0 + wv * 16 + n;          // A-matrix row (edge)
  v8f h0 = {}, h1 = {}, h2 = {}, h3 = {};
#pragma unroll
  for (int s = 0; s < 4; ++s) {
    const int kb = 4 * s + 2 * hi;
    v2f a = ld2(ES + eg * 16 + kb);
    v2f b0 = { Wfc0[kb * 64 + n],      Wfc0[(kb + 1) * 64 + n] };
    v2f b1 = { Wfc0[kb * 64 + 16 + n], Wfc0[(kb + 1) * 64 + 16 + n] };
    v2f b2 = { Wfc0[kb * 64 + 32 + n], Wfc0[(kb + 1) * 64 + 32 + n] };
    v2f b3 = { Wfc0[kb * 64 + 48 + n], Wfc0[(kb + 1) * 64 + 48 + n] };
    h0 = wmma4(a, b0, h0);
    h1 = wmma4(a, b1, h1);
    h2 = wmma4(a, b2, h2);
    h3 = wmma4(a, b3, h3);
  }
  float* hb = &hbuf[wv * (16 * 68)];
#pragma unroll
  for (int r = 0; r < 8; ++r) {
    const int m = 8 * hi + r;
    float x;
    x = h0[r] * 0.25f; hb[m * 68 + n]      = (x / (1.0f + __expf(-x))) * (SILU_G * INV8);
    x = h1[r] * 0.25f; hb[m * 68 + 16 + n] = (x / (1.0f + __expf(-x))) * (SILU_G * INV8);
    x = h2[r] * 0.25f; hb[m * 68 + 32 + n] = (x / (1.0f + __expf(-x))) * (SILU_G * INV8);
    x = h3[r] * 0.25f; hb[m * 68 + 48 + n] = (x / (1.0f + __expf(-x))) * (SILU_G * INV8);
  }
  __syncthreads();

  // ---- GEMM2: w = h @ Wfc1 (K=64 -> 16 steps, 10 column tiles of 16)
  v8f acc[10] = {};
#pragma unroll
  for (int s = 0; s < 16; ++s) {
    const int kb = 4 * s + 2 * hi;
    v2f a = *(const v2f*)(hb + n * 68 + kb);
#pragma unroll
    for (int t = 0; t < 10; ++t) {
      v2f b = { Wfc1[kb * 160 + 16 * t + n], Wfc1[(kb + 1) * 160 + 16 * t + n] };
      acc[t] = wmma4(a, b, acc[t]);
    }
  }

  // ---- elementwise tensor product + scatter (atomics land in L2-resident acc)
  const int wb = wv * 16;
#pragma unroll
  for (int r = 0; r < 8; ++r) {
    const int m = wb + 8 * hi + r;
    const int src = sSrc[m], dst = sDst[m];
    const float ea0 = sEA[m * 4 + 0];
    const float ev0 = sEA[m * 4 + 1], ev1 = sEA[m * 4 + 2], ev2 = sEA[m * 4 + 3];
#pragma unroll
    for (int t = 0; t < 2; ++t) {
      const int u = 16 * t + n;
      const float w0 = acc[0 + t][r], w1 = acc[2 + t][r], w2 = acc[4 + t][r],
                  w3 = acc[6 + t][r], w4 = acc[8 + t][r];
      const float xs  = Ls[src * 32 + u];
      const float xv0 = Lv[src * 96 + u * 3 + 0];
      const float xv1 = Lv[src * 96 + u * 3 + 1];
      const float xv2 = Lv[src * 96 + u * 3 + 2];
      const float mid0 = w0 * xs * ea0;
      const float dotv = xv0 * ev0 + xv1 * ev1 + xv2 * ev2;
      const float mid1 = w3 * dotv * INV_SQ3;
      const float w1xs = QTR * w1 * xs;
      const float cr0 = xv1 * ev2 - xv2 * ev1;
      const float cr1 = xv2 * ev0 - xv0 * ev2;
      const float cr2 = xv0 * ev1 - xv1 * ev0;
      atomAddF(&AccS[dst * 64 + u],      QTR * mid0);
      atomAddF(&AccS[dst * 64 + 32 + u], QTR * mid1);
      float* av = AccV + (size_t)dst * 288;
      atomAddF(&av[u * 3 + 0], w1xs * ev0);
      atomAddF(&av[u * 3 + 1], w1xs * ev1);
      atomAddF(&av[u * 3 + 2], w1xs * ev2);
      const float w2e = QTR * w2 * ea0;
      atomAddF(&av[(32 + u) * 3 + 0], w2e * xv0);
      atomAddF(&av[(32 + u) * 3 + 1], w2e * xv1);
      atomAddF(&av[(32 + u) * 3 + 2], w2e * xv2);
      const float w4s = QTR * w4 * INV_SQ2;
      atomAddF(&av[(64 + u) * 3 + 0], w4s * cr0);
      atomAddF(&av[(64 + u) * 3 + 1], w4s * cr1);
      atomAddF(&av[(64 + u) * 3 + 2], w4s * cr2);
    }
  }
}

// ---------------------------------------------------------------- node post
__global__ void node_post(const float* __restrict__ X, const float* __restrict__ NA,
                          const float* __restrict__ Wsc0, const float* __restrict__ Wsc1,
                          const float* __restrict__ Wl20, const float* __restrict__ Wl21,
                          const float* __restrict__ AccS, const float* __restrict__ AccV,
                          float* __restrict__ Out) {
  const int lane = threadIdx.x & 31, wv = threadIdx.x >> 5;
  const int tile = blockIdx.x * 8 + wv;
  if (tile * 16 >= N_NODES) return;               // wave-uniform
  const int n = lane & 15, hi = lane >> 4;
  const int arow = tile * 16 + n;
  float nam[8];
#pragma unroll
  for (int r = 0; r < 8; ++r) nam[r] = NA[tile * 16 + 8 * hi + r];

  // s_s = xs @ W_sc0 ;  x2_s = node_s @ W_l20
  v8f ss0 = {}, ss1 = {};
#pragma unroll
  for (int s = 0; s < 8; ++s) {
    const int kb = 4 * s + 2 * hi;
    v2f a = ld2(X + arow * 128 + kb);
    v2f b0 = { Wsc0[kb * 32 + n],      Wsc0[(kb + 1) * 32 + n] };
    v2f b1 = { Wsc0[kb * 32 + 16 + n], Wsc0[(kb + 1) * 32 + 16 + n] };
    ss0 = wmma4(a, b0, ss0);
    ss1 = wmma4(a, b1, ss1);
  }
  v8f x20 = {}, x21 = {};
#pragma unroll
  for (int s = 0; s < 16; ++s) {
    const int kb = 4 * s + 2 * hi;
    v2f a = ld2(AccS + (size_t)arow * 64 + kb);
    v2f b0 = { Wl20[kb * 32 + n],      Wl20[(kb + 1) * 32 + n] };
    v2f b1 = { Wl20[kb * 32 + 16 + n], Wl20[(kb + 1) * 32 + 16 + n] };
    x20 = wmma4(a, b0, x20);
    x21 = wmma4(a, b1, x21);
  }
#pragma unroll
  for (int r = 0; r < 8; ++r) {
    const int node = tile * 16 + 8 * hi + r;
    Out[node * 128 + n]      = (CC_S * F032 * ss0[r] + CC_X * INV8 * x20[r]) * nam[r];
    Out[node * 128 + 16 + n] = (CC_S * F032 * ss1[r] + CC_X * INV8 * x21[r]) * nam[r];
  }

  // vector channels
#pragma unroll
  for (int i = 0; i < 3; ++i) {
    v8f sv0 = {}, sv1 = {};
#pragma unroll
    for (int s = 0; s < 8; ++s) {
      const int kb = 4 * s + 2 * hi;
      v2f a = { X[arow * 128 + 32 + kb * 3 + i], X[arow * 128 + 32 + (kb + 1) * 3 + i] };
      v2f b0 = { Wsc1[kb * 32 + n],      Wsc1[(kb + 1) * 32 + n] };
      v2f b1 = { Wsc1[kb * 32 + 16 + n], Wsc1[(kb + 1) * 32 + 16 + n] };
      sv0 = wmma4(a, b0, sv0);
      sv1 = wmma4(a, b1, sv1);
    }
    v8f v20 = {}, v21 = {};
#pragma unroll
    for (int s = 0; s < 24; ++s) {
      const int kb = 4 * s + 2 * hi;
      v2f a = { AccV[(size_t)arow * 288 + kb * 3 + i],
                AccV[(size_t)arow * 288 + (kb + 1) * 3 + i] };
      v2f b0 = { Wl21[kb * 32 + n],      Wl21[(kb + 1) * 32 + n] };
      v2f b1 = { Wl21[kb * 32 + 16 + n], Wl21[(kb + 1) * 32 + 16 + n] };
      v20 = wmma4(a, b0, v20);
      v21 = wmma4(a, b1, v21);
    }
#pragma unroll
    for (int r = 0; r < 8; ++r) {
      const int node = tile * 16 + 8 * hi + r;
      Out[node * 128 + 32 + n * 3 + i] =
          (CC_S * F032 * sv0[r] + CC_X * INV_SQ96 * v20[r]) * nam[r];
      Out[node * 128 + 32 + (16 + n) * 3 + i] =
          (CC_S * F032 * sv1[r] + CC_X * INV_SQ96 * v21[r]) * nam[r];
    }
  }
}

// ---------------------------------------------------------------- launcher
extern "C" void kernel_launch(void* const* d_in, const int* in_sizes, int n_in,
                              void* d_out, int out_size, void* d_ws, size_t ws_size,
                              hipStream_t stream) {
  (void)in_sizes; (void)n_in; (void)out_size; (void)ws_size;
  const float* node_input = (const float*)d_in[0];
  const float* node_attr  = (const float*)d_in[1];
  const int*   edge_index = (const int*)d_in[2];
  const float* edge_attr  = (const float*)d_in[3];
  const float* edge_scal  = (const float*)d_in[4];
  const float* W_sc0 = (const float*)d_in[5];
  const float* W_sc1 = (const float*)d_in[6];
  const float* W_l10 = (const float*)d_in[7];
  const float* W_l11 = (const float*)d_in[8];
  const float* Wfc0  = (const float*)d_in[9];
  const float* Wfc1  = (const float*)d_in[10];
  const float* W_l20 = (const float*)d_in[11];
  const float* W_l21 = (const float*)d_in[12];

  float* ws   = (float*)d_ws;
  float* Ls   = ws;                         // N*32
  float* Lv   = ws + (size_t)N_NODES * 32;  // N*96
  float* AccS = ws + (size_t)N_NODES * 128; // N*64
  float* AccV = ws + (size_t)N_NODES * 192; // N*288  (contiguous after AccS)
  float* Out  = (float*)d_out;

  const int nz = N_NODES * 352;             // AccS + AccV contiguous
  zero_f32<<<(nz + 255) / 256, 256, 0, stream>>>(AccS, nz);

  const int node_blocks = (N_NODES / 16 + 7) / 8;  // 2500 tiles, 8 waves/block
  node_pre<<<node_blocks, 256, 0, stream>>>(node_input, node_attr, W_l10, W_l11, Ls, Lv);

  edge_kernel<<<N_EDGES / 128, 256, 0, stream>>>(edge_scal, edge_index, edge_attr,
                                                 Wfc0, Wfc1, Ls, Lv, AccS, AccV);

  node_post<<<node_blocks, 256, 0, stream>>>(node_input, node_attr, W_sc0, W_sc1,
                                             W_l20, W_l21, AccS, AccV, Out);
}